// GRNN_257698038354
// MI455X (gfx1250) — compile-verified
//
#include <hip/hip_runtime.h>
#include <cstddef>
#include <cstdint>

#define NODES 4096
#define DIM   512
#define HID   512
#define NEDGE 131072
#define NREL  3

#define BM 128
#define BN 128
#define BK 16

// LDS staging geometry (floats): A tile 128x16 padded to 17/row, B tile padded
// to 132/row (NN) or 17/row (NT). Both fit in 2176 floats.
#define ABUF  2176           // 128*17
#define BBUF  2176           // >= 16*132 and 128*17
#define STAGE (ABUF + BBUF)  // one double-buffer stage

typedef float v2f __attribute__((ext_vector_type(2)));
typedef float v8f __attribute__((ext_vector_type(8)));
typedef unsigned int u32x4 __attribute__((ext_vector_type(4)));
typedef int i32x4 __attribute__((ext_vector_type(4)));
typedef int i32x8 __attribute__((ext_vector_type(8)));

// ---------------------------------------------------------------------------
// TDM: issue one 2D tensor_load_to_lds.
//   lds_byte   : wave-relative LDS byte address of tile destination
//   gaddr      : global byte address of tile start
//   d0 / d1    : tile dim0 (contiguous, elements) / dim1 (rows)
//   stride0    : row stride in elements
//   pad_iv/amt : encoded pad_interval (2^(c+1) dwords) / pad_amount (c+1 dwords)
// tensor_dim == tile_dim, so no OOB path is exercised.
// ---------------------------------------------------------------------------
__device__ __forceinline__ void tdm_load_2d(unsigned lds_byte,
                                            unsigned long long gaddr,
                                            unsigned d0, unsigned d1,
                                            unsigned long long stride0,
                                            unsigned pad_iv, unsigned pad_amt)
{
    u32x4 g0;
    g0.x = 1u;                                   // count=1, user descriptor
    g0.y = lds_byte;                             // lds_addr [63:32]
    g0.z = (unsigned)(gaddr & 0xffffffffull);    // global_addr[31:0]
    g0.w = (unsigned)((gaddr >> 32) & 0x01ffffffull) | 0x80000000u; // type=2

    i32x8 g1;
    g1[0] = (int)((2u << 16) | (1u << 20) | (pad_iv << 22) | (pad_amt << 25));
    g1[1] = (int)((d0 & 0xffffu) << 16);                              // tensor_dim0 lo16
    g1[2] = (int)(((d0 >> 16) & 0xffffu) | ((d1 & 0xffffu) << 16));   // dim0 hi / dim1 lo
    g1[3] = (int)(((d1 >> 16) & 0xffffu) | ((d0 & 0xffffu) << 16));   // dim1 hi / tile_dim0
    g1[4] = (int)(d1 & 0xffffu);                                      // tile_dim1 (tile_dim2=0)
    g1[5] = (int)(stride0 & 0xffffffffull);                           // dim0_stride lo32
    g1[6] = (int)((stride0 >> 32) & 0xffffull);                       // dim0_stride hi16
    g1[7] = 0;

    i32x4 z4 = {0, 0, 0, 0};
#if defined(__clang_major__) && (__clang_major__ >= 23)
    i32x8 z8 = {0, 0, 0, 0, 0, 0, 0, 0};
    __builtin_amdgcn_tensor_load_to_lds(g0, g1, z4, z4, z8, 0);
#else
    __builtin_amdgcn_tensor_load_to_lds(g0, g1, z4, z4, 0);
#endif
}

__device__ __forceinline__ unsigned lds_byte_addr(const void* p)
{
    // LDS aperture: low 32 bits of a flat shared address are the wave-relative
    // LDS byte offset.
    return (unsigned)(uintptr_t)p;
}

// ---------------------------------------------------------------------------
// WMMA f32 GEMM with TDM-staged, double-buffered LDS tiles.
// C[M,N] = A[M,K] * op(B);  epilog: 0 plain, 1 zero-diag, 2 add bias[col]
// All dims must be multiples of the tile sizes (true for this problem).
// ---------------------------------------------------------------------------
__global__ __launch_bounds__(256) void gemm_f32_wmma(
    const float* __restrict__ A, const float* __restrict__ B,
    float* __restrict__ C, int M, int N, int K,
    int lda, int ldb, int ldc, int transB, int epilog,
    const float* __restrict__ bias)
{
    __shared__ float smem[2 * STAGE];

    const int t     = threadIdx.x;
    const int lane  = t & 31;
    const int w     = t >> 5;
    const int wm    = (w & 3) * 32;   // wave row offset within 128x128 tile
    const int wn    = (w >> 2) * 64;  // wave col offset
    const int bm    = blockIdx.y * BM;
    const int bn    = blockIdx.x * BN;
    const int lhalf = lane >> 4;
    const int lq    = lane & 15;

    v8f acc[2][4];
    const v8f zero8 = {0.f, 0.f, 0.f, 0.f, 0.f, 0.f, 0.f, 0.f};
#pragma unroll
    for (int i = 0; i < 2; ++i)
#pragma unroll
        for (int j = 0; j < 4; ++j) acc[i][j] = zero8;

    const unsigned ldsbase = lds_byte_addr(smem);
    const int nch = K / BK;

    // prologue: DMA chunk 0 into stage 0 (single wave issues; TDM ignores EXEC)
    if (w == 0) {
        unsigned al = ldsbase;
        unsigned bl = ldsbase + ABUF * 4u;
        tdm_load_2d(al, (unsigned long long)(uintptr_t)(A) +
                        ((unsigned long long)bm * lda) * 4ull,
                    BK, BM, (unsigned long long)lda, 3u, 0u);
        if (!transB)
            tdm_load_2d(bl, (unsigned long long)(uintptr_t)(B) +
                            ((unsigned long long)bn) * 4ull,
                        BN, BK, (unsigned long long)ldb, 6u, 3u);
        else
            tdm_load_2d(bl, (unsigned long long)(uintptr_t)(B) +
                            ((unsigned long long)bn * ldb) * 4ull,
                        BK, BN, (unsigned long long)ldb, 3u, 0u);
    }

    for (int chunk = 0; chunk < nch; ++chunk) {
        const int stage = chunk & 1;
        float* Abuf = smem + stage * STAGE;
        float* Bbuf = Abuf + ABUF;

        if (w == 0) {
            if (chunk + 1 < nch) {
                const int k1 = (chunk + 1) * BK;
                const int st = stage ^ 1;
                unsigned al = ldsbase + (unsigned)(st * STAGE) * 4u;
                unsigned bl = al + ABUF * 4u;
                tdm_load_2d(al, (unsigned long long)(uintptr_t)(A) +
                                ((unsigned long long)bm * lda + k1) * 4ull,
                            BK, BM, (unsigned long long)lda, 3u, 0u);
                if (!transB)
                    tdm_load_2d(bl, (unsigned long long)(uintptr_t)(B) +
                                    ((unsigned long long)k1 * ldb + bn) * 4ull,
                                BN, BK, (unsigned long long)ldb, 6u, 3u);
                else
                    tdm_load_2d(bl, (unsigned long long)(uintptr_t)(B) +
                                    ((unsigned long long)bn * ldb + k1) * 4ull,
                                BK, BN, (unsigned long long)ldb, 3u, 0u);
                __builtin_amdgcn_s_wait_tensorcnt(2);   // current stage complete
            } else {
                __builtin_amdgcn_s_wait_tensorcnt(0);   // last stage complete
            }
        }
        __syncthreads();

        // ---- 4 WMMA K-steps of 4 over this 16-wide chunk
#pragma unroll
        for (int ks = 0; ks < 4; ++ks) {
            const int kb = ks * 4 + lhalf * 2;  // lanes 0-15: K=0,1 ; 16-31: K=2,3
            v2f af[2], bf[4];
#pragma unroll
            for (int i = 0; i < 2; ++i) {
                int r = wm + i * 16 + lq;
                af[i].x = Abuf[r * 17 + kb];
                af[i].y = Abuf[r * 17 + kb + 1];
            }
            if (!transB) {
#pragma unroll
                for (int j = 0; j < 4; ++j) {
                    int c = wn + j * 16 + lq;
                    bf[j].x = Bbuf[kb * 132 + c];
                    bf[j].y = Bbuf[(kb + 1) * 132 + c];
                }
            } else {
#pragma unroll
                for (int j = 0; j < 4; ++j) {
                    int c = wn + j * 16 + lq;
                    bf[j].x = Bbuf[c * 17 + kb];
                    bf[j].y = Bbuf[c * 17 + kb + 1];
                }
            }
#pragma unroll
            for (int i = 0; i < 2; ++i)
#pragma unroll
                for (int j = 0; j < 4; ++j)
                    acc[i][j] = __builtin_amdgcn_wmma_f32_16x16x4_f32(
                        false, af[i], false, bf[j], (short)0, acc[i][j],
                        false, false);
        }
        __syncthreads();   // stage consumed; next iteration may DMA over it
    }

    // ---- epilogue (C/D layout: VGPR v, lane l -> row v+8*(l>>4), col l&15)
#pragma unroll
    for (int i = 0; i < 2; ++i)
#pragma unroll
        for (int j = 0; j < 4; ++j)
#pragma unroll
            for (int v = 0; v < 8; ++v) {
                int grow = bm + wm + i * 16 + v + lhalf * 8;
                int gcol = bn + wn + j * 16 + lq;
                float val = acc[i][j][v];
                if (epilog == 1 && grow == gcol) val = 0.0f;
                if (epilog == 2) val += bias[gcol];
                C[(size_t)grow * ldc + gcol] = val;
            }
}

// ---------------------------------------------------------------------------
// Row normalize: p[row][:] /= (sum==0 ? 1 : sum)
// ---------------------------------------------------------------------------
__global__ void row_normalize(float* __restrict__ Mtx, int n)
{
    __shared__ float red[256];
    const int row = blockIdx.x, t = threadIdx.x;
    float* p = Mtx + (size_t)row * n;
    float s = 0.f;
    for (int j = t; j < n; j += 256) s += p[j];
    red[t] = s; __syncthreads();
    for (int st = 128; st; st >>= 1) { if (t < st) red[t] += red[t + st]; __syncthreads(); }
    float d = red[0];
    if (d == 0.f) d = 1.f;
    const float inv = 1.f / d;
    for (int j = t; j < n; j += 256) p[j] *= inv;
}

// ---------------------------------------------------------------------------
// Per-row k-th largest threshold via 32-step binary search over LDS row cache
// ---------------------------------------------------------------------------
__global__ void row_topk_thresh(const float* __restrict__ P, float* __restrict__ thr,
                                int n, int k)
{
    __shared__ float rowbuf[NODES];
    __shared__ float red[256];
    const int row = blockIdx.x, t = threadIdx.x;
    const float* p = P + (size_t)row * n;
    float mn = 3.4e38f, mx = -3.4e38f;
    for (int j = t; j < n; j += 256) {
        float v = p[j]; rowbuf[j] = v;
        mn = fminf(mn, v); mx = fmaxf(mx, v);
    }
    red[t] = mx; __syncthreads();
    for (int s = 128; s; s >>= 1) { if (t < s) red[t] = fmaxf(red[t], red[t + s]); __syncthreads(); }
    mx = red[0]; __syncthreads();
    red[t] = mn; __syncthreads();
    for (int s = 128; s; s >>= 1) { if (t < s) red[t] = fminf(red[t], red[t + s]); __syncthreads(); }
    mn = red[0]; __syncthreads();

    float lo = mn, hi = mx;
    for (int it = 0; it < 32; ++it) {
        float mid = 0.5f * (lo + hi);
        int c = 0;
        for (int j = t; j < n; j += 256) c += (rowbuf[j] >= mid) ? 1 : 0;
        red[t] = (float)c; __syncthreads();
        for (int s = 128; s; s >>= 1) { if (t < s) red[t] += red[t + s]; __syncthreads(); }
        int tot = (int)red[0]; __syncthreads();
        if (tot >= k) lo = mid; else hi = mid;
    }
    if (t == 0) thr[row] = lo;
}

// P[i][j] = P[i][j] >= thr[i] ? P[i][j] : 0
__global__ void mask_rows(float* __restrict__ P, const float* __restrict__ thr)
{
    size_t i = (size_t)blockIdx.x * 256 + threadIdx.x;
    int row = (int)(i >> 12);   // / 4096
    float v = P[i];
    P[i] = (v >= thr[row]) ? v : 0.f;
}

// ---------------------------------------------------------------------------
// RGCN edge scatter (per relation): counts and messages (f32 atomics)
// ---------------------------------------------------------------------------
__global__ void scatter_cnt(float* __restrict__ cnt,
                            const int* __restrict__ dst, const int* __restrict__ et,
                            const unsigned char* __restrict__ keep, int r)
{
    int e = blockIdx.x * 256 + threadIdx.x;
    if (e >= NEDGE) return;
    if (et[e] == r && keep[e]) atomicAdd(&cnt[dst[e]], 1.0f);
}

__global__ void scatter_msg(const float* __restrict__ T, float* __restrict__ S,
                            const int* __restrict__ src, const int* __restrict__ dst,
                            const int* __restrict__ et,
                            const unsigned char* __restrict__ keep, int r)
{
    long idx = (long)blockIdx.x * 256 + threadIdx.x;
    int e = (int)(idx >> 9);       // / 512
    int c = (int)(idx & 511);
    if (et[e] != r || !keep[e]) return;
    atomicAdd(&S[(size_t)dst[e] * HID + c], T[(size_t)src[e] * HID + c]);
}

// out += S / max(cnt, 1)
__global__ void combine_mean(float* __restrict__ out, const float* __restrict__ S,
                             const float* __restrict__ cnt)
{
    size_t idx = (size_t)blockIdx.x * 256 + threadIdx.x;
    int row = (int)(idx >> 9);
    out[idx] += S[idx] / fmaxf(cnt[row], 1.0f);
}

// ---------------------------------------------------------------------------
// Fused ReLU + LayerNorm + dropout(keep=0.4) per row (H = 512)
// ---------------------------------------------------------------------------
__global__ void relu_ln_drop(const float* __restrict__ in, float* __restrict__ out,
                             const float* __restrict__ w, const float* __restrict__ b,
                             const unsigned char* __restrict__ drop)
{
    __shared__ float red[256];
    const int row = blockIdx.x, t = threadIdx.x;
    const float* xr = in + (size_t)row * HID;
    float v0 = fmaxf(xr[t], 0.f), v1 = fmaxf(xr[t + 256], 0.f);
    red[t] = v0 + v1; __syncthreads();
    for (int s = 128; s; s >>= 1) { if (t < s) red[t] += red[t + s]; __syncthreads(); }
    float mu = red[0] * (1.f / HID); __syncthreads();
    float d0 = v0 - mu, d1 = v1 - mu;
    red[t] = d0 * d0 + d1 * d1; __syncthreads();
    for (int s = 128; s; s >>= 1) { if (t < s) red[t] += red[t + s]; __syncthreads(); }
    float inv = rsqrtf(red[0] * (1.f / HID) + 1e-5f);
    size_t base = (size_t)row * HID;
    float y0 = d0 * inv * w[t] + b[t];
    float y1 = d1 * inv * w[t + 256] + b[t + 256];
    out[base + t]       = drop[base + t]       ? y0 * 2.5f : 0.f;
    out[base + t + 256] = drop[base + t + 256] ? y1 * 2.5f : 0.f;
}

// ---------------------------------------------------------------------------
// Binary adjacency build (idempotent store handles duplicate edges like max)
// ---------------------------------------------------------------------------
__global__ void scatter_A(float* __restrict__ A, const int* __restrict__ src,
                          const int* __restrict__ dst,
                          const unsigned char* __restrict__ keep)
{
    int e = blockIdx.x * 256 + threadIdx.x;
    if (e >= NEDGE) return;
    if (keep[e]) A[(size_t)src[e] * NODES + dst[e]] = 1.0f;
}

// h = 0.9 h + 0.1 t
__global__ void push_axpy(float* __restrict__ h, const float* __restrict__ t)
{
    size_t i = (size_t)blockIdx.x * 256 + threadIdx.x;
    h[i] = 0.9f * h[i] + 0.1f * t[i];
}

// ---------------------------------------------------------------------------
extern "C" void kernel_launch(void* const* d_in, const int* in_sizes, int n_in,
                              void* d_out, int out_size, void* d_ws, size_t ws_size,
                              hipStream_t stream)
{
    const float* x       = (const float*)d_in[0];
    const float* W1_rel  = (const float*)d_in[1];
    const float* W1_root = (const float*)d_in[2];
    const float* b1      = (const float*)d_in[3];
    const float* W2_rel  = (const float*)d_in[4];
    const float* W2_root = (const float*)d_in[5];
    const float* b2      = (const float*)d_in[6];
    const float* ln1w = (const float*)d_in[7];
    const float* ln1b = (const float*)d_in[8];
    const float* ln2w = (const float*)d_in[9];
    const float* ln2b = (const float*)d_in[10];
    const int* eidx  = (const int*)d_in[11];
    const int* etype = (const int*)d_in[12];
    const unsigned char* keep  = (const unsigned char*)d_in[13];
    const unsigned char* drop1 = (const unsigned char*)d_in[14];
    const unsigned char* drop2 = (const unsigned char*)d_in[15];
    const int* src = eidx;
    const int* dst = eidx + NEDGE;

    const size_t NN = (size_t)NODES * NODES;       // 16.7M
    const size_t NH = (size_t)NODES * HID;         // 2.1M
    float* base = (float*)d_ws;
    float* P    = base;             // [N,N]
    float* Aadj = P + NN;           // [N,N]
    float* B0   = Aadj + NN;        // [N,H]  h
    float* B1   = B0 + NH;          // [N,H]  layer out accumulator
    float* B2   = B1 + NH;          // [N,H]  per-relation transform / push tmp
    float* B3   = B2 + NH;          // [N,H]  per-relation scatter sums
    float* thr  = B3 + NH;          // [N]
    float* cnt  = thr + NODES;      // [N]

    const dim3 blk(256);
    const dim3 gridNN(NODES / BN, NODES / BM);   // 32x32
    const dim3 gridNH(HID / BN, NODES / BM);     // 4x32

    // 1) P = x x^T with zero diagonal; row-normalize; top-k(2048) thresholds
    gemm_f32_wmma<<<gridNN, blk, 0, stream>>>(x, x, P, NODES, NODES, DIM,
                                              DIM, DIM, NODES, 1, 1, nullptr);
    row_normalize<<<NODES, 256, 0, stream>>>(P, NODES);
    row_topk_thresh<<<NODES, 256, 0, stream>>>(P, thr, NODES, NODES / 2);

    // 2) RGCN layer 1 (input x)
    gemm_f32_wmma<<<gridNH, blk, 0, stream>>>(x, W1_root, B1, NODES, HID, DIM,
                                              DIM, HID, HID, 0, 2, b1);
    for (int r = 0; r < NREL; ++r) {
        gemm_f32_wmma<<<gridNH, blk, 0, stream>>>(x, W1_rel + (size_t)r * DIM * HID,
                                                  B2, NODES, HID, DIM,
                                                  DIM, HID, HID, 0, 0, nullptr);
        hipMemsetAsync(B3, 0, NH * sizeof(float), stream);
        hipMemsetAsync(cnt, 0, NODES * sizeof(float), stream);
        scatter_cnt<<<NEDGE / 256, 256, 0, stream>>>(cnt, dst, etype, keep, r);
        scatter_msg<<<(unsigned)((size_t)NEDGE * HID / 256), 256, 0, stream>>>(
            B2, B3, src, dst, etype, keep, r);
        combine_mean<<<(unsigned)(NH / 256), 256, 0, stream>>>(B1, B3, cnt);
    }
    relu_ln_drop<<<NODES, 256, 0, stream>>>(B1, B0, ln1w, ln1b, drop1);

    // 3) RGCN layer 2 (input B0)
    gemm_f32_wmma<<<gridNH, blk, 0, stream>>>(B0, W2_root, B1, NODES, HID, HID,
                                              HID, HID, HID, 0, 2, b2);
    for (int r = 0; r < NREL; ++r) {
        gemm_f32_wmma<<<gridNH, blk, 0, stream>>>(B0, W2_rel + (size_t)r * HID * HID,
                                                  B2, NODES, HID, HID,
                                                  HID, HID, HID, 0, 0, nullptr);
        hipMemsetAsync(B3, 0, NH * sizeof(float), stream);
        hipMemsetAsync(cnt, 0, NODES * sizeof(float), stream);
        scatter_cnt<<<NEDGE / 256, 256, 0, stream>>>(cnt, dst, etype, keep, r);
        scatter_msg<<<(unsigned)((size_t)NEDGE * HID / 256), 256, 0, stream>>>(
            B2, B3, src, dst, etype, keep, r);
        combine_mean<<<(unsigned)(NH / 256), 256, 0, stream>>>(B1, B3, cnt);
    }
    relu_ln_drop<<<NODES, 256, 0, stream>>>(B1, B0, ln2w, ln2b, drop2);

    // 4) forward_push: binary kept-edge adjacency, row-normalized; 3 iterations
    hipMemsetAsync(Aadj, 0, NN * sizeof(float), stream);
    scatter_A<<<NEDGE / 256, 256, 0, stream>>>(Aadj, src, dst, keep);
    row_normalize<<<NODES, 256, 0, stream>>>(Aadj, NODES);
    for (int it = 0; it < 3; ++it) {
        gemm_f32_wmma<<<gridNH, blk, 0, stream>>>(Aadj, B0, B2, NODES, HID, NODES,
                                                  NODES, HID, HID, 0, 0, nullptr);
        push_axpy<<<(unsigned)(NH / 256), 256, 0, stream>>>(B0, B2);
    }

    // 5) sparsify P and final out = Psp @ h
    mask_rows<<<(unsigned)(NN / 256), 256, 0, stream>>>(P, thr);
    gemm_f32_wmma<<<gridNH, blk, 0, stream>>>(P, B0, (float*)d_out, NODES, HID, NODES,
                                              NODES, HID, HID, 0, 0, nullptr);
}